// MLPdirected_57226144252655
// MI455X (gfx1250) — compile-verified
//
#include <hip/hip_runtime.h>

#define N_NODE 4096
#define FMAPS  64
#define HIDDEN 8
#define ROWCHUNKS 16                 // colsum reduction stages
#define ROWS_PER_CHUNK (N_NODE / ROWCHUNKS)

typedef float v2f __attribute__((ext_vector_type(2)));
typedef float v4f __attribute__((ext_vector_type(4)));
typedef float v8f __attribute__((ext_vector_type(8)));

// ---------------------------------------------------------------------------
// Stage 1a: partial column sums of A.  grid (16,16) x 256 threads.
// Each block: 256 columns x 256 rows, coalesced 1KB rows, NT loads (A streamed once).
// ---------------------------------------------------------------------------
__global__ void colsum_partial_kernel(const float* __restrict__ A, float* __restrict__ part) {
    const int j  = blockIdx.x * blockDim.x + threadIdx.x;   // column
    const int r0 = blockIdx.y * ROWS_PER_CHUNK;
    float s = 0.0f;
    #pragma unroll 8
    for (int i = 0; i < ROWS_PER_CHUNK; ++i) {
        s += __builtin_nontemporal_load(&A[(size_t)(r0 + i) * N_NODE + j]);
    }
    part[blockIdx.y * N_NODE + j] = s;
}

// ---------------------------------------------------------------------------
// Stage 1b: finalize sum_w[j] = (colsum[j] - A[j,j]) / (N-1).  16 x 256.
// ---------------------------------------------------------------------------
__global__ void colsum_finalize_kernel(const float* __restrict__ part,
                                       const float* __restrict__ A,
                                       float* __restrict__ sum_w) {
    const int j = blockIdx.x * blockDim.x + threadIdx.x;
    float s = 0.0f;
    #pragma unroll
    for (int c = 0; c < ROWCHUNKS; ++c) s += part[c * N_NODE + j];
    s -= A[(size_t)j * N_NODE + j];
    sum_w[j] = s * (1.0f / (float)(N_NODE - 1));
}

// ---------------------------------------------------------------------------
// Stage 2: fused small GEMM via V_WMMA_F32_16X16X4_F32 (wave32, one wave/block).
//   D(16 nodes x 16) = E_tile(16x64) @ Wcat^T(64x16), K-loop of 16 steps.
//   Wcat rows 0..7  = W1[:, :64]   -> aj (add b1)
//   Wcat rows 8..15 = W1[:, 64:128]-> bk (add sum_w * W1[:,128])
// Fragment layouts per CDNA5 ISA 7.12.2:
//   A 16x4 f32 : lanes g=lane>>4, m=lane&15 -> M=m, VGPR0/1 = K = 2g / 2g+1
//   B  4x16    : N = lane&15, VGPR0/1 = K rows 2g / 2g+1 (symmetric striping)
//   C/D 16x16  : N = lane&15, VGPR r -> M = r + 8g
// ---------------------------------------------------------------------------
__global__ void small_mlp_wmma_kernel(const float* __restrict__ emb,   // (64,4096) row-major
                                      const float* __restrict__ W1,    // (8,129)
                                      const float* __restrict__ b1,    // (8)
                                      const float* __restrict__ sum_w, // (4096)
                                      float* __restrict__ aj,          // (4096,8)
                                      float* __restrict__ bk) {        // (4096,8)
    const int n0   = blockIdx.x * 16;
    const int lane = threadIdx.x;       // 0..31
    const int g    = lane >> 4;
    const int m    = lane & 15;

    v8f c = {};
    #pragma unroll
    for (int kk = 0; kk < 16; ++kk) {
        const int f0 = kk * 4 + 2 * g;  // K rows held by this lane group
        v2f a, b;
        // A fragment: E[n0+m, f0], E[n0+m, f0+1]  (E[n,f] = emb[f*N + n])
        a.x = emb[(size_t)f0 * N_NODE + n0 + m];
        a.y = emb[(size_t)(f0 + 1) * N_NODE + n0 + m];
        // B fragment: Wcat[m, f0], Wcat[m, f0+1]
        if (m < 8) {
            b.x = W1[m * 129 + f0];
            b.y = W1[m * 129 + f0 + 1];
        } else {
            b.x = W1[(m - 8) * 129 + 64 + f0];
            b.y = W1[(m - 8) * 129 + 64 + f0 + 1];
        }
        c = __builtin_amdgcn_wmma_f32_16x16x4_f32(
                /*neg_a=*/false, a, /*neg_b=*/false, b,
                /*c_mod=*/(short)0, c, /*reuse_a=*/false, /*reuse_b=*/false);
    }

    #pragma unroll
    for (int r = 0; r < 8; ++r) {
        const int node = n0 + r + 8 * g;
        if (m < 8) {
            aj[node * HIDDEN + m] = c[r] + b1[m];
        } else {
            const int hb = m - 8;
            bk[node * HIDDEN + hb] = c[r] + sum_w[node] * W1[hb * 129 + 128];
        }
    }
}

// ---------------------------------------------------------------------------
// Stage 3: fused streaming output (the 64MB write).
//   out[j,k] = b2[0] + sum_h relu(aj[j,h] + bk[k,h]) * W2[0,h]
// Block: 256 threads, each thread owns 4 consecutive columns (float4 NT store).
// bk for those columns lives in registers; aj[j,:] is block-uniform (scalar loads).
// Hidden dim handled as float2 pairs to encourage v_pk_add_f32 / v_pk_fma_f32.
// grid = (4096/1024, 4096/32) = (4, 128).
// ---------------------------------------------------------------------------
#define OUT_ROWS 32
__global__ void fuse_out_kernel(const float* __restrict__ aj,
                                const float* __restrict__ bk,
                                const float* __restrict__ W2,  // (8,8); row 0 used
                                const float* __restrict__ b2,  // (8); [0] used
                                float* __restrict__ out) {
    const int c0 = blockIdx.x * (256 * 4) + threadIdx.x * 4;
    const int r0 = blockIdx.y * OUT_ROWS;

    v2f w2p[4];
    #pragma unroll
    for (int p = 0; p < 4; ++p) { w2p[p].x = W2[2 * p]; w2p[p].y = W2[2 * p + 1]; }
    const float bias = b2[0];

    // bk fragments for this thread's 4 columns: 4 x 8 floats in registers
    v2f bkp[4][4];
    #pragma unroll
    for (int i = 0; i < 4; ++i) {
        const float* src = &bk[(size_t)(c0 + i) * HIDDEN];
        #pragma unroll
        for (int p = 0; p < 4; ++p) { bkp[i][p].x = src[2 * p]; bkp[i][p].y = src[2 * p + 1]; }
    }

    for (int j = r0; j < r0 + OUT_ROWS; ++j) {
        v2f ajp[4];
        const float* asrc = &aj[(size_t)j * HIDDEN];   // block-uniform -> scalar loads
        #pragma unroll
        for (int p = 0; p < 4; ++p) { ajp[p].x = asrc[2 * p]; ajp[p].y = asrc[2 * p + 1]; }

        v4f o;
        #pragma unroll
        for (int i = 0; i < 4; ++i) {
            v2f acc2 = {0.0f, 0.0f};
            #pragma unroll
            for (int p = 0; p < 4; ++p) {
                v2f t = ajp[p] + bkp[i][p];           // v_pk_add_f32
                t.x = fmaxf(t.x, 0.0f);               // relu (no pk max f32)
                t.y = fmaxf(t.y, 0.0f);
                acc2 += t * w2p[p];                   // v_pk_fma_f32
            }
            o[i] = acc2.x + acc2.y + bias;
        }
        __builtin_nontemporal_store(o, (v4f*)&out[(size_t)j * N_NODE + c0]);
    }
}

// ---------------------------------------------------------------------------
// Launch: A colsum (2 stages) -> WMMA small MLP -> fused streaming output.
// ws layout (floats): part[16*4096] | sum_w[4096] | aj[4096*8] | bk[4096*8]
// Total ~528 KB; everything overwritten each call (deterministic, no atomics).
// ---------------------------------------------------------------------------
extern "C" void kernel_launch(void* const* d_in, const int* in_sizes, int n_in,
                              void* d_out, int out_size, void* d_ws, size_t ws_size,
                              hipStream_t stream) {
    const float* A   = (const float*)d_in[0];  // (1,4096,4096)
    const float* emb = (const float*)d_in[1];  // (1,64,4096)
    const float* W1  = (const float*)d_in[2];  // (8,129)
    const float* b1  = (const float*)d_in[3];  // (8)
    const float* W2  = (const float*)d_in[4];  // (8,8)
    const float* b2  = (const float*)d_in[5];  // (8)
    float* out = (float*)d_out;

    float* ws    = (float*)d_ws;
    float* part  = ws;                              // 65536
    float* sumw  = ws + (size_t)ROWCHUNKS * N_NODE; // 4096
    float* aj    = sumw + N_NODE;                   // 32768
    float* bk    = aj + (size_t)N_NODE * HIDDEN;    // 32768

    colsum_partial_kernel<<<dim3(N_NODE / 256, ROWCHUNKS), 256, 0, stream>>>(A, part);
    colsum_finalize_kernel<<<N_NODE / 256, 256, 0, stream>>>(part, A, sumw);
    small_mlp_wmma_kernel<<<N_NODE / 16, 32, 0, stream>>>(emb, W1, b1, sumw, aj, bk);
    fuse_out_kernel<<<dim3(N_NODE / 1024, N_NODE / OUT_ROWS), 256, 0, stream>>>(aj, bk, W2, b2, out);
}